// VLI_82952998355465
// MI455X (gfx1250) — compile-verified
//
#include <hip/hip_runtime.h>
#include <hip/hip_bf16.h>

// Problem constants (match reference)
#define NB    16384   // B
#define NP    32      // P
#define ND    128     // D_LANE == D_TRAJ == D_INT
#define ROWS  128     // rows of flattened (B*P, D) handled per block (= 4 batches)
#define PITCH 132     // LDS row pitch in floats (bank-conflict padding)

typedef __attribute__((ext_vector_type(2))) float v2f;
typedef __attribute__((ext_vector_type(8))) float v8f;

// Load a 128x128 row-major weight matrix into LDS *transposed* with pitch
// PITCH: sWT[i*PITCH + d] = w[d*ND + i]. B-fragments then read contiguous
// (k, k+1) pairs per lane -> ds_load_b64 straight into aligned VGPR pairs.
__device__ __forceinline__ void stage_weight_T(const float* __restrict__ w,
                                               float* __restrict__ sWT, int tid) {
    const float4* w4 = (const float4*)w;
    #pragma unroll
    for (int i = 0; i < 16; ++i) {
        int q  = i * 256 + tid;      // float4 index, 4096 total
        int d  = q >> 5;             // k-dim row of w
        int i0 = (q & 31) * 4;       // n-dim col
        float4 x = w4[q];
        sWT[(i0 + 0) * PITCH + d] = x.x;
        sWT[(i0 + 1) * PITCH + d] = x.y;
        sWT[(i0 + 2) * PITCH + d] = x.z;
        sWT[(i0 + 3) * PITCH + d] = x.w;
    }
}

// ---------------------------------------------------------------------------
// Kernel 1: G[b,i] = sum_d agent[b,d] * w2[d,i]   (B x 128) @ (128 x 128)
// ---------------------------------------------------------------------------
__global__ __launch_bounds__(256) void vli_gemm_g(const float* __restrict__ agent,
                                                  const float* __restrict__ w2,
                                                  float* __restrict__ g) {
    extern __shared__ float smem[];
    float* sA  = smem;                // ROWS x PITCH
    float* sWT = sA + ROWS * PITCH;   // 128 x PITCH (w2 transposed: [i][d])

    const int tid  = threadIdx.x;
    const int row0 = blockIdx.x * ROWS;

    const float4* a4 = (const float4*)(agent + (size_t)row0 * ND);
    #pragma unroll
    for (int i = 0; i < 16; ++i) {
        int q = i * 256 + tid;
        int r = q >> 5, c4 = q & 31;
        *(float4*)(sA + r * PITCH + c4 * 4) = a4[q];
    }
    stage_weight_T(w2, sWT, tid);
    __syncthreads();

    const int wave = tid >> 5;
    const int lane = tid & 31;
    const int hi   = lane >> 4;
    const int l16  = lane & 15;
    const int mrow = wave * 16;

    v8f acc[8] = {};
    const float* aBase = sA  + (mrow + l16) * PITCH + 2 * hi;
    const float* bBase = sWT + l16 * PITCH + 2 * hi;
    #pragma unroll 4
    for (int kk = 0; kk < 32; ++kk) {
        const int k0 = kk * 4;
        v2f a = *(const v2f*)(aBase + k0);
        #pragma unroll
        for (int tn = 0; tn < 8; ++tn) {
            v2f b = *(const v2f*)(bBase + tn * 16 * PITCH + k0);
            acc[tn] = __builtin_amdgcn_wmma_f32_16x16x4_f32(
                false, a, false, b, (short)0, acc[tn], false, false);
        }
    }

    #pragma unroll
    for (int tn = 0; tn < 8; ++tn) {
        const int col = tn * 16 + l16;
        #pragma unroll
        for (int r = 0; r < 8; ++r) {
            g[(size_t)(row0 + mrow + r + 8 * hi) * ND + col] = acc[tn][r];
        }
    }
}

// ---------------------------------------------------------------------------
// Kernel 2: fused additive attention (4 batches per block, one HBM pass).
// ---------------------------------------------------------------------------
__global__ __launch_bounds__(256) void vli_attn(const float* __restrict__ lanes,
                                                const float* __restrict__ w1,
                                                const float* __restrict__ g,
                                                const float* __restrict__ v,
                                                float* __restrict__ out) {
    extern __shared__ float smem[];
    float* sL     = smem;                  // ROWS x PITCH   (lane tile)
    float* sWT    = sL + ROWS * PITCH;     // 128 x PITCH    (w1 transposed [i][d])
    float* sG     = sWT + ND * PITCH;      // 4 x 128
    float* sV     = sG + 4 * ND;           // 128
    float* sScore = sV + ND;               // 128
    float* sProb  = sScore + ROWS;         // 128
    float* sAtt   = sProb + ROWS;          // 4 x 128

    const int tid    = threadIdx.x;
    const int rowG0  = blockIdx.x * ROWS;
    const int batch0 = blockIdx.x * 4;

    // ---- stage inputs ----
    const float4* l4g = (const float4*)(lanes + (size_t)rowG0 * ND);
    #pragma unroll
    for (int i = 0; i < 16; ++i) {
        int q = i * 256 + tid;
        int r = q >> 5, c4 = q & 31;
        *(float4*)(sL + r * PITCH + c4 * 4) = l4g[q];
    }
    stage_weight_T(w1, sWT, tid);
    if (tid < 128) ((float4*)sG)[tid] = ((const float4*)(g + (size_t)batch0 * ND))[tid];
    if (tid < 32)  ((float4*)sV)[tid] = ((const float4*)v)[tid];
    __syncthreads();

    const int wave = tid >> 5;
    const int lane = tid & 31;
    const int hi   = lane >> 4;
    const int l16  = lane & 15;
    const int mrow = wave * 16;
    const int bLoc = wave >> 1;

    // ---- GEMM: H-tile = L[mrow..mrow+15, :] @ W1 ----
    v8f acc[8] = {};
    const float* aBase = sL  + (mrow + l16) * PITCH + 2 * hi;
    const float* bBase = sWT + l16 * PITCH + 2 * hi;
    #pragma unroll 4
    for (int kk = 0; kk < 32; ++kk) {
        const int k0 = kk * 4;
        v2f a = *(const v2f*)(aBase + k0);
        #pragma unroll
        for (int tn = 0; tn < 8; ++tn) {
            v2f b = *(const v2f*)(bBase + tn * 16 * PITCH + k0);
            acc[tn] = __builtin_amdgcn_wmma_f32_16x16x4_f32(
                false, a, false, b, (short)0, acc[tn], false, false);
        }
    }

    // ---- tanh + dot with v, in registers ----
    float s[8] = {0.f, 0.f, 0.f, 0.f, 0.f, 0.f, 0.f, 0.f};
    #pragma unroll
    for (int tn = 0; tn < 8; ++tn) {
        const int col = tn * 16 + l16;
        const float gadd = sG[bLoc * ND + col];
        const float vv   = sV[col];
        #pragma unroll
        for (int r = 0; r < 8; ++r) {
            s[r] += tanhf(acc[tn][r] + gadd) * vv;
        }
    }
    // reduce the 16 lanes of each half (xor masks stay within the half)
    #pragma unroll
    for (int r = 0; r < 8; ++r) {
        float x = s[r];
        x += __shfl_xor(x, 1);
        x += __shfl_xor(x, 2);
        x += __shfl_xor(x, 4);
        x += __shfl_xor(x, 8);
        if (l16 == r) sScore[mrow + r + 8 * hi] = x;   // row = mrow + r + 8*hi
    }
    __syncthreads();

    // ---- softmax over P=32, one wave per batch ----
    if (wave < 4) {
        float sc = sScore[wave * 32 + lane];
        float mx = sc;
        #pragma unroll
        for (int m = 16; m >= 1; m >>= 1) mx = fmaxf(mx, __shfl_xor(mx, m));
        float e = __expf(sc - mx);
        float sum = e;
        #pragma unroll
        for (int m = 16; m >= 1; m >>= 1) sum += __shfl_xor(sum, m);
        sProb[wave * 32 + lane] = e / sum;
    }
    __syncthreads();

    // ---- att[b,d] = sum_p prob[b,p] * L[b,p,d] ----
    #pragma unroll
    for (int pass = 0; pass < 2; ++pass) {
        const int idx = pass * 256 + tid;
        const int bb = idx >> 7, d = idx & 127;
        float a = 0.f;
        #pragma unroll 8
        for (int p = 0; p < 32; ++p)
            a += sProb[bb * 32 + p] * sL[(bb * 32 + p) * PITCH + d];
        sAtt[idx] = a;
    }
    __syncthreads();

    // ---- out = att - L * prob, coalesced float4 stores ----
    float4* o4 = (float4*)(out + (size_t)rowG0 * ND);
    #pragma unroll
    for (int i = 0; i < 16; ++i) {
        int q = i * 256 + tid;
        int r = q >> 5, c4 = q & 31;
        int bb = r >> 5;
        const float pr = sProb[r];
        const float4 l4 = *(const float4*)(sL + r * PITCH + c4 * 4);
        const float4 a4 = *(const float4*)(sAtt + bb * ND + c4 * 4);
        float4 o;
        o.x = a4.x - l4.x * pr;
        o.y = a4.y - l4.y * pr;
        o.z = a4.z - l4.z * pr;
        o.w = a4.w - l4.w * pr;
        o4[q] = o;
    }
}

extern "C" void kernel_launch(void* const* d_in, const int* in_sizes, int n_in,
                              void* d_out, int out_size, void* d_ws, size_t ws_size,
                              hipStream_t stream) {
    const float* agent = (const float*)d_in[0];   // [B, 128]
    const float* lanes = (const float*)d_in[1];   // [B, P, 128]
    const float* w1    = (const float*)d_in[2];   // [128, 128]
    const float* w2    = (const float*)d_in[3];   // [128, 128]
    const float* v     = (const float*)d_in[4];   // [128, 1]
    float* out = (float*)d_out;                   // [B, P, 128]
    float* g   = (float*)d_ws;                    // [B, 128] scratch (8.4 MB)

    const size_t smem1 = (size_t)(ROWS * PITCH + ND * PITCH) * sizeof(float);
    vli_gemm_g<<<NB / ROWS, 256, smem1, stream>>>(agent, w2, g);

    const size_t smem2 = (size_t)(ROWS * PITCH + ND * PITCH + 4 * ND + ND +
                                  ROWS + ROWS + 4 * ND) * sizeof(float);
    vli_attn<<<(NB * NP) / ROWS, 256, smem2, stream>>>(lanes, w1, g, v, out);
}